// densityPropGRUCell_35940286333595
// MI455X (gfx1250) — compile-verified
//
#include <hip/hip_runtime.h>
#include <cmath>

typedef float v2f __attribute__((ext_vector_type(2)));
typedef float v8f __attribute__((ext_vector_type(8)));

constexpr int Bn  = 64;
constexpr int Un  = 256;
constexpr int UUn = Un * Un;

// ---------------------------------------------------------------- utilities
__device__ inline float blk_reduce(float v, float* red) {
  const int t = threadIdx.x;
  red[t] = v;
  __syncthreads();
  for (int s = 128; s > 0; s >>= 1) {
    if (t < s) red[t] += red[t + s];
    __syncthreads();
  }
  const float r = red[0];
  __syncthreads();
  return r;
}

__device__ inline v8f wmma4(v2f a, v2f b, v8f c) {
  return __builtin_amdgcn_wmma_f32_16x16x4_f32(false, a, false, b, (short)0, c,
                                               false, false);
}

// Mainloop: acc[t] += A[b](i0..i0+15, :) x W(:, j0+16t..)   (C = A @ W)
__device__ inline void loop_AxW(const float* __restrict__ Ab,
                                const float* __restrict__ W,
                                int i0, int j0, int lr, int half, v8f acc[4]) {
#pragma unroll 4
  for (int k0 = 0; k0 < Un; k0 += 4) {
    const int kb = k0 + 2 * half;
    v2f a;
    a.x = Ab[(i0 + lr) * Un + kb];
    a.y = Ab[(i0 + lr) * Un + kb + 1];
#pragma unroll
    for (int t = 0; t < 4; ++t) {
      v2f bb;
      bb.x = W[kb * Un + j0 + 16 * t + lr];
      bb.y = W[(kb + 1) * Un + j0 + 16 * t + lr];
      acc[t] = wmma4(a, bb, acc[t]);
    }
  }
}

// Mainloop: acc[t] += W^T(i0..,:) x A[b](:, j0+16t..)   (C = W^T @ A)
__device__ inline void loop_WtX(const float* __restrict__ W,
                                const float* __restrict__ Ab,
                                int i0, int j0, int lr, int half, v8f acc[4]) {
#pragma unroll 4
  for (int k0 = 0; k0 < Un; k0 += 4) {
    const int kb = k0 + 2 * half;
    v2f a;
    a.x = W[kb * Un + i0 + lr];          // (W^T)[i,k] = W[k,i]
    a.y = W[(kb + 1) * Un + i0 + lr];
#pragma unroll
    for (int t = 0; t < 4; ++t) {
      v2f bb;
      bb.x = Ab[kb * Un + j0 + 16 * t + lr];
      bb.y = Ab[(kb + 1) * Un + j0 + 16 * t + lr];
      acc[t] = wmma4(a, bb, acc[t]);
    }
  }
}

// ---------------------------------------------------------------- softplus of the 6 sigma vectors
__global__ __launch_bounds__(256) void softplus_kernel(
    const float* __restrict__ wz, const float* __restrict__ uz,
    const float* __restrict__ wr, const float* __restrict__ ur,
    const float* __restrict__ wh, const float* __restrict__ uh,
    float* __restrict__ sp) {
  const int u = threadIdx.x;
  sp[0 * Un + u] = log1pf(expf(wz[u]));
  sp[1 * Un + u] = log1pf(expf(uz[u]));
  sp[2 * Un + u] = log1pf(expf(wr[u]));
  sp[3 * Un + u] = log1pf(expf(ur[u]));
  sp[4 * Un + u] = log1pf(expf(wh[u]));
  sp[5 * Un + u] = log1pf(expf(uh[u]));
}

// ---------------------------------------------------------------- gates + per-batch scalars + mu_out
__global__ __launch_bounds__(256) void gates_kernel(
    const float* __restrict__ inputs, const float* __restrict__ prev,
    const float* __restrict__ Sigma,
    const float* __restrict__ Uz, const float* __restrict__ Wz,
    const float* __restrict__ Ur, const float* __restrict__ Wr,
    const float* __restrict__ Uh, const float* __restrict__ Wh,
    float* __restrict__ z, float* __restrict__ r, float* __restrict__ h,
    float* __restrict__ gz, float* __restrict__ gr, float* __restrict__ gh,
    float* __restrict__ xsq, float* __restrict__ psq,
    float* __restrict__ gsq, float* __restrict__ trS, float* __restrict__ mu_out) {
  __shared__ float xin[Un], pin[Un], gin[Un], red[256];
  const int b = blockIdx.x, u = threadIdx.x;
  const float xv = inputs[b * Un + u];
  const float pv = prev[b * Un + u];
  xin[u] = xv;
  pin[u] = pv;
  __syncthreads();

  float az = 0.f, ar = 0.f;
  for (int d = 0; d < Un; ++d) {
    const float x = xin[d], p = pin[d];
    az = fmaf(x, Uz[d * Un + u], az);
    az = fmaf(p, Wz[d * Un + u], az);
    ar = fmaf(x, Ur[d * Un + u], ar);
    ar = fmaf(p, Wr[d * Un + u], ar);
  }
  const float zv = 1.f / (1.f + expf(-az));
  const float rv = 1.f / (1.f + expf(-ar));
  const float gv = pv * rv;
  gin[u] = gv;
  __syncthreads();

  float ah = 0.f;
  for (int d = 0; d < Un; ++d) {
    ah = fmaf(xin[d], Uh[d * Un + u], ah);
    ah = fmaf(gin[d], Wh[d * Un + u], ah);
  }
  const float hv = tanhf(ah);

  z[b * Un + u]  = zv;
  r[b * Un + u]  = rv;
  h[b * Un + u]  = hv;
  gz[b * Un + u] = zv * (1.f - zv);
  gr[b * Un + u] = rv * (1.f - rv);
  gh[b * Un + u] = 1.f - hv * hv;
  mu_out[b * Un + u] = zv * pv + (1.f - zv) * hv;

  const float sx  = blk_reduce(xv * xv, red);
  const float sp2 = blk_reduce(pv * pv, red);
  const float sg  = blk_reduce(gv * gv, red);
  const float st  = blk_reduce(Sigma[(size_t)b * UUn + (size_t)u * Un + u], red);
  if (u == 0) { xsq[b] = sx; psq[b] = sp2; gsq[b] = sg; trS[b] = st; }
}

// ---------------------------------------------------------------- C[b] = A[b] @ W  (wave computes 16x64 strip)
__global__ __launch_bounds__(32) void gemm_SxW(
    const float* __restrict__ A, const float* __restrict__ W,
    float* __restrict__ C) {
  const int lane = threadIdx.x;
  const int half = lane >> 4, lr = lane & 15;
  const int j0 = blockIdx.x * 64, i0 = blockIdx.y * 16, b = blockIdx.z;
  const float* Ab = A + (size_t)b * UUn;
  v8f acc[4] = {};
  loop_AxW(Ab, W, i0, j0, lr, half, acc);
  float* Cb = C + (size_t)b * UUn;
#pragma unroll
  for (int t = 0; t < 4; ++t)
#pragma unroll
    for (int v = 0; v < 8; ++v)
      Cb[(i0 + v + 8 * half) * Un + j0 + 16 * t + lr] = acc[t][v];
}

// ---------------------------------------------------------------- Sigma_z = epi(W^T @ A[b]) : staged to d_out
__global__ __launch_bounds__(32) void gemm_WtX_epi(
    const float* __restrict__ W, const float* __restrict__ A, float* __restrict__ Cout,
    const float* __restrict__ gate, const float* __restrict__ s1,
    const float* __restrict__ s2, const float* __restrict__ xsq,
    const float* __restrict__ spw, const float* __restrict__ spu) {
  const int lane = threadIdx.x;
  const int half = lane >> 4, lr = lane & 15;
  const int j0 = blockIdx.x * 64, i0 = blockIdx.y * 16, b = blockIdx.z;
  const float* Ab = A + (size_t)b * UUn;
  v8f acc[4] = {};
  loop_WtX(W, Ab, i0, j0, lr, half, acc);
  const float ds = s1[b] + s2[b];
  const float xb = xsq[b];
  float* Cb = Cout + (size_t)b * UUn;
#pragma unroll
  for (int t = 0; t < 4; ++t) {
    const int col = j0 + 16 * t + lr;
    const float gj = gate[b * Un + col];
#pragma unroll
    for (int v = 0; v < 8; ++v) {
      const int row = i0 + v + 8 * half;
      float x = acc[t][v];
      if (row == col) x += ds * spw[row] + xb * spu[row];
      Cb[row * Un + col] = x * gate[b * Un + row] * gj;
    }
  }
}

// ---------------------------------------------------------------- sigma_g fused: out = Ss*Sr + (p⊗p)Sr + (r⊗r)Ss,
// where Sr = epi(W_r^T @ A[b]) is formed in-register
__global__ __launch_bounds__(32) void gemm_WtX_epi_g(
    const float* __restrict__ W, const float* __restrict__ A, float* __restrict__ Cout,
    const float* __restrict__ gate, const float* __restrict__ s1,
    const float* __restrict__ s2, const float* __restrict__ xsq,
    const float* __restrict__ spw, const float* __restrict__ spu,
    const float* __restrict__ Ss, const float* __restrict__ prev,
    const float* __restrict__ r) {
  const int lane = threadIdx.x;
  const int half = lane >> 4, lr = lane & 15;
  const int j0 = blockIdx.x * 64, i0 = blockIdx.y * 16, b = blockIdx.z;
  const float* Ab = A + (size_t)b * UUn;
  v8f acc[4] = {};
  loop_WtX(W, Ab, i0, j0, lr, half, acc);
  const float dsv = s1[b] + s2[b];
  const float xb = xsq[b];
  const float* Ssb = Ss + (size_t)b * UUn;
  float* Cb = Cout + (size_t)b * UUn;
#pragma unroll
  for (int t = 0; t < 4; ++t) {
    const int col = j0 + 16 * t + lr;
    const float gj = gate[b * Un + col];
    const float pj = prev[b * Un + col];
    const float rj = r[b * Un + col];
#pragma unroll
    for (int v = 0; v < 8; ++v) {
      const int row = i0 + v + 8 * half;
      float x = acc[t][v];
      if (row == col) x += dsv * spw[row] + xb * spu[row];
      const float sr = x * gate[b * Un + row] * gj;       // Sigma_out_r element
      const float ss = Ssb[row * Un + col];
      const float pi = prev[b * Un + row];
      const float ri = r[b * Un + row];
      Cb[row * Un + col] = ss * sr + pi * pj * sr + ri * rj * ss;  // sigma_g
    }
  }
}

// ---------------------------------------------------------------- final fused: Sigma_h formed in-register, then
// full output combination written in place over Sigma_z staged in d_out
__global__ __launch_bounds__(32) void gemm_WtX_epi_final(
    const float* __restrict__ W, const float* __restrict__ A,
    float* __restrict__ So /* in: Sigma_z, out: Sigma_out */,
    const float* __restrict__ gate /* gh */, const float* __restrict__ s1,
    const float* __restrict__ s2, const float* __restrict__ xsq,
    const float* __restrict__ spw, const float* __restrict__ spu,
    const float* __restrict__ Ss, const float* __restrict__ z,
    const float* __restrict__ prev, const float* __restrict__ h) {
  const int lane = threadIdx.x;
  const int half = lane >> 4, lr = lane & 15;
  const int j0 = blockIdx.x * 64, i0 = blockIdx.y * 16, b = blockIdx.z;
  const float* Ab = A + (size_t)b * UUn;
  v8f acc[4] = {};
  loop_WtX(W, Ab, i0, j0, lr, half, acc);
  const float dsv = s1[b] + s2[b];
  const float xb = xsq[b];
  const float* Ssb = Ss + (size_t)b * UUn;
  float* Sob = So + (size_t)b * UUn;
#pragma unroll
  for (int t = 0; t < 4; ++t) {
    const int col = j0 + 16 * t + lr;
    const float gj = gate[b * Un + col];
    const float zj = z[b * Un + col];
    const float pj = prev[b * Un + col];
    const float hj = h[b * Un + col];
#pragma unroll
    for (int v = 0; v < 8; ++v) {
      const int row = i0 + v + 8 * half;
      float x = acc[t][v];
      if (row == col) x += dsv * spw[row] + xb * spu[row];
      const float sh = x * gate[b * Un + row] * gj;       // Sigma_out_h element
      const size_t idx = (size_t)row * Un + col;
      const float sz = Sob[idx];                          // Sigma_out_z (staged)
      const float ss = Ssb[idx];
      const float zi = z[b * Un + row];
      const float pi = prev[b * Un + row];
      const float hi = h[b * Un + row];
      const float sa = sz * ss + zi * zj * ss + pi * pj * sz;
      const float sb = sz * sh + (1.f - zi) * (1.f - zj) * sh + hi * hj * sz;
      float o = sa + sb - sz * (pi * hj + hi * pj);
      if (!(fabsf(o) < __builtin_inff())) o = 0.0f;       // NaN or Inf -> 0
      if (row == col) o = fabsf(o);
      Sob[idx] = o;
    }
  }
}

// ---------------------------------------------------------------- trace of sigma_g
__global__ __launch_bounds__(256) void trace_kernel(const float* __restrict__ Sg,
                                                    float* __restrict__ trG) {
  __shared__ float red[256];
  const int b = blockIdx.x, i = threadIdx.x;
  const float v = blk_reduce(Sg[(size_t)b * UUn + (size_t)i * Un + i], red);
  if (i == 0) trG[b] = v;
}

// ---------------------------------------------------------------- launcher
extern "C" void kernel_launch(void* const* d_in, const int* in_sizes, int n_in,
                              void* d_out, int out_size, void* d_ws, size_t ws_size,
                              hipStream_t stream) {
  const float* inputs = (const float*)d_in[0];
  const float* prev   = (const float*)d_in[1];
  const float* Sigma  = (const float*)d_in[2];
  const float* Uz  = (const float*)d_in[3];
  const float* uzs = (const float*)d_in[4];
  const float* Wz  = (const float*)d_in[5];
  const float* wzs = (const float*)d_in[6];
  const float* Ur  = (const float*)d_in[7];
  const float* urs = (const float*)d_in[8];
  const float* Wr  = (const float*)d_in[9];
  const float* wrs = (const float*)d_in[10];
  const float* Uh  = (const float*)d_in[11];
  const float* uhs = (const float*)d_in[12];
  const float* Wh  = (const float*)d_in[13];
  const float* whs = (const float*)d_in[14];

  float* out    = (float*)d_out;
  float* mu_out = out;              // [B,U]
  float* SigOut = out + Bn * Un;    // [B,U,U] — also stages Sigma_z

  float* ws   = (float*)d_ws;
  float* buf0 = ws;                        // [B,U,U] GEMM intermediate
  float* buf1 = buf0 + (size_t)Bn * UUn;   // [B,U,U] sigma_g
  float* zP   = buf1 + (size_t)Bn * UUn;
  float* rP   = zP + Bn * Un;
  float* hP   = rP + Bn * Un;
  float* gzP  = hP + Bn * Un;
  float* grP  = gzP + Bn * Un;
  float* ghP  = grP + Bn * Un;
  float* xsqP = ghP + Bn * Un;
  float* psqP = xsqP + Bn;
  float* gsqP = psqP + Bn;
  float* trSP = gsqP + Bn;
  float* trGP = trSP + Bn;
  float* spP  = trGP + Bn;          // 6 softplus vectors of U

  softplus_kernel<<<1, Un, 0, stream>>>(wzs, uzs, wrs, urs, whs, uhs, spP);
  gates_kernel<<<Bn, Un, 0, stream>>>(inputs, prev, Sigma, Uz, Wz, Ur, Wr, Uh, Wh,
                                      zP, rP, hP, gzP, grP, ghP,
                                      xsqP, psqP, gsqP, trSP, mu_out);

  dim3 gw(Un / 64, Un / 16, Bn), bw(32);
  // --- z quad: Sigma_z staged into d_out's Sigma region
  gemm_SxW<<<gw, bw, 0, stream>>>(Sigma, Wz, buf0);
  gemm_WtX_epi<<<gw, bw, 0, stream>>>(Wz, buf0, SigOut, gzP, psqP, trSP, xsqP,
                                      spP + 0 * Un, spP + 1 * Un);
  // --- r quad fused with sigma_g -> buf1
  gemm_SxW<<<gw, bw, 0, stream>>>(Sigma, Wr, buf0);
  gemm_WtX_epi_g<<<gw, bw, 0, stream>>>(Wr, buf0, buf1, grP, psqP, trSP, xsqP,
                                        spP + 2 * Un, spP + 3 * Un,
                                        Sigma, prev, rP);
  // --- trace of sigma_g
  trace_kernel<<<Bn, Un, 0, stream>>>(buf1, trGP);
  // --- h quad fused with final combination (in place over d_out Sigma region)
  gemm_SxW<<<gw, bw, 0, stream>>>(buf1, Wh, buf0);
  gemm_WtX_epi_final<<<gw, bw, 0, stream>>>(Wh, buf0, SigOut, ghP, gsqP, trGP, xsqP,
                                            spP + 4 * Un, spP + 5 * Un,
                                            Sigma, zP, prev, hP);
}